// BoundaryAwareLoss_83863531422201
// MI455X (gfx1250) — compile-verified
//
#include <hip/hip_runtime.h>
#include <math.h>

typedef __attribute__((ext_vector_type(2))) float v2f;
typedef __attribute__((ext_vector_type(8))) float v8f;

#define B_   4
#define H_   352
#define W_   352
#define HW_  (H_ * W_)
#define NT_  (B_ * HW_)          // 495616 = 7744 * 64
#define CHUNKS_PER_B 1936        // HW_ / 64 (exact)
#define INF_ 1e10f
#define SIGMA_INV 0.2f
#define LAMBDA_ 0.5f
#define TC 16                    // columns per tile in the column pass
#define NWAVE 968                // 7744 chunks / 8 chunks-per-wave

// ---------------------------------------------------------------- init
__global__ void init_mm_kernel(unsigned* __restrict__ mm) {
    int t = threadIdx.x;
    if (t < B_)                mm[t] = 0x7F800000u;  // +inf bits (running min of w>=0)
    if (t >= B_ && t < 2 * B_) mm[t] = 0u;           // 0.0f bits (running max of w>=0)
}

// ---------------------------------------------------------------- pass 1: min-plus along W (rows)
__global__ void row_minplus_kernel(const float* __restrict__ target,
                                   float* __restrict__ r_bg,
                                   float* __restrict__ r_fg) {
    __shared__ float gb[W_];
    __shared__ float gf[W_];
    const int row = blockIdx.x;                      // 0 .. B*H-1
    const float* trow = target + (size_t)row * W_;
    for (int k = threadIdx.x; k < W_; k += blockDim.x) {
        float m = (trow[k] != 0.0f) ? 1.0f : 0.0f;
        gb[k] = m * INF_;            // g for EDT(mask): INF at foreground
        gf[k] = (1.0f - m) * INF_;   // g for EDT(~mask): INF at background
    }
    __syncthreads();
    for (int x = threadIdx.x; x < W_; x += blockDim.x) {
        float db = INF_, df = INF_;
        float d = (float)x;                          // carried displacement, no cvt in loop
        #pragma unroll 4
        for (int k = 0; k < W_; ++k) {
            float sq = d * d;
            db = fminf(db, sq + gb[k]);
            df = fminf(df, sq + gf[k]);
            d -= 1.0f;
        }
        r_bg[(size_t)row * W_ + x] = db;
        r_fg[(size_t)row * W_ + x] = df;
    }
}

// ---------------------------------------------------------------- pass 2: min-plus along H (cols) + w + per-sample min/max
// Tile staging uses CDNA5 async global->LDS DMA (ASYNCcnt-tracked, no VGPR round-trip).
__global__ void col_minplus_w_kernel(const float* __restrict__ r_bg,
                                     const float* __restrict__ r_fg,
                                     float* __restrict__ w,          // may alias r_bg (disjoint LDS-staged tiles)
                                     unsigned* __restrict__ mm) {
    __shared__ float sb[H_ * TC];
    __shared__ float sf[H_ * TC];
    __shared__ float red[256];
    const int b  = blockIdx.x / (W_ / TC);
    const int x0 = (blockIdx.x % (W_ / TC)) * TC;
    const size_t base = (size_t)b * HW_;
    const float* gbp = r_bg + base;
    const float* gfp = r_fg + base;

    for (int flat = threadIdx.x; flat < H_ * TC; flat += blockDim.x) {
        int j = flat / TC, c = flat % TC;
        unsigned boff = (unsigned)(((unsigned)j * W_ + x0 + c) * 4u);  // byte offset within sample
        unsigned lb = (unsigned)(uintptr_t)&sb[flat];                  // LDS byte address (low 32 bits)
        unsigned lf = (unsigned)(uintptr_t)&sf[flat];
        asm volatile("global_load_async_to_lds_b32 %0, %1, %2"
                     :: "v"(lb), "v"(boff), "s"(gbp) : "memory");
        asm volatile("global_load_async_to_lds_b32 %0, %1, %2"
                     :: "v"(lf), "v"(boff), "s"(gfp) : "memory");
    }
    asm volatile("s_wait_asynccnt 0" ::: "memory");
    __syncthreads();

    float lmin = INF_, lmax = 0.0f;
    for (int flat = threadIdx.x; flat < H_ * TC; flat += blockDim.x) {
        int y = flat / TC, c = flat % TC;
        float db = INF_, df = INF_;
        float d = (float)y;
        #pragma unroll 4
        for (int j = 0; j < H_; ++j) {
            float sq = d * d;
            db = fminf(db, sq + sb[j * TC + c]);
            df = fminf(df, sq + sf[j * TC + c]);
            d -= 1.0f;
        }
        float dist = sqrtf(db) + sqrtf(df);          // |sdt| = dist_bg + dist_fg (disjoint support)
        float wv   = __expf(-dist * SIGMA_INV);
        w[base + (size_t)y * W_ + x0 + c] = wv;
        lmin = fminf(lmin, wv);
        lmax = fmaxf(lmax, wv);
    }

    red[threadIdx.x] = lmin; __syncthreads();
    for (int s = 128; s > 0; s >>= 1) {
        if (threadIdx.x < s) red[threadIdx.x] = fminf(red[threadIdx.x], red[threadIdx.x + s]);
        __syncthreads();
    }
    if (threadIdx.x == 0) atomicMin(&mm[b], __float_as_uint(red[0]));
    __syncthreads();
    red[threadIdx.x] = lmax; __syncthreads();
    for (int s = 128; s > 0; s >>= 1) {
        if (threadIdx.x < s) red[threadIdx.x] = fmaxf(red[threadIdx.x], red[threadIdx.x + s]);
        __syncthreads();
    }
    if (threadIdx.x == 0) atomicMax(&mm[B_ + b], __float_as_uint(red[0]));
}

// ---------------------------------------------------------------- hoist per-sample normalization constants
__global__ void finalize_mm_kernel(const unsigned* __restrict__ mm, float* __restrict__ mmf) {
    int b = threadIdx.x;
    if (b < B_) {
        float wmin = __uint_as_float(mm[b]);
        float wmax = __uint_as_float(mm[B_ + b]);
        mmf[b]      = wmin;
        mmf[B_ + b] = 1.0f / (wmax - wmin + 1e-6f);  // one precise divide per sample
    }
}

// ---------------------------------------------------------------- weighted BCE contribution
// softplus(-|p|) via v_exp_f32 + v_log_f32: 1+e in (1,2], abs error < 2^-24 only where the
// term itself is < 2^-24 of a 0.7-mean sum — negligible, and keeps this kernel memory-bound.
__device__ __forceinline__ float contribf(int idx, float wmin, float winv,
                                          const float* __restrict__ pred,
                                          const float* __restrict__ target,
                                          const float* __restrict__ w) {
    float p  = pred[idx];
    float t  = target[idx];
    float wn = (w[idx] - wmin) * winv;
    float bce = fmaxf(p, 0.0f) - p * t + __logf(1.0f + __expf(-fabsf(p)));
    return bce * (1.0f + LAMBDA_ * wn);
}

// ---------------------------------------------------------------- WMMA sum reduction (one partial per wave)
// A = ones(16x4): D = A*B + C  =>  sum(D) = 16 * sum(B) independent of B's lane layout.
__global__ void loss_wmma_kernel(const float* __restrict__ pred,
                                 const float* __restrict__ target,
                                 const float* __restrict__ w,
                                 const float* __restrict__ mmf,
                                 float* __restrict__ part) {
    const int lane = threadIdx.x & 31;
    const int wave = (blockIdx.x * blockDim.x + threadIdx.x) >> 5;   // 0 .. NWAVE-1

    v2f ones; ones.x = 1.0f; ones.y = 1.0f;
    v8f acc = {};
    #pragma unroll
    for (int it = 0; it < 8; ++it) {
        int chunk = wave * 8 + it;                   // 7744 chunks of 64 elements, exact cover
        int b = chunk / CHUNKS_PER_B;                // chunks never straddle samples
        float wmin = mmf[b];
        float winv = mmf[B_ + b];
        int i0 = chunk * 64 + lane;
        v2f bv;
        bv.x = contribf(i0,      wmin, winv, pred, target, w);
        bv.y = contribf(i0 + 32, wmin, winv, pred, target, w);
        acc = __builtin_amdgcn_wmma_f32_16x16x4_f32(false, ones, false, bv,
                                                    (short)0, acc, false, false);
    }
    float p = acc[0] + acc[1] + acc[2] + acc[3] + acc[4] + acc[5] + acc[6] + acc[7];
    #pragma unroll
    for (int off = 16; off > 0; off >>= 1) p += __shfl_xor(p, off, 32);
    if (lane == 0) part[wave] = p;
}

// ---------------------------------------------------------------- deterministic final reduction
__global__ void final_kernel(const float* __restrict__ part, float* __restrict__ out) {
    __shared__ float red[256];
    float s = 0.0f;
    for (int i = threadIdx.x; i < NWAVE; i += 256) s += part[i];
    red[threadIdx.x] = s; __syncthreads();
    for (int st = 128; st > 0; st >>= 1) {
        if (threadIdx.x < st) red[threadIdx.x] += red[threadIdx.x + st];
        __syncthreads();
    }
    if (threadIdx.x == 0)
        out[0] = red[0] * (1.0f / 16.0f) * (1.0f / (float)NT_);  // /16 ones-matmul fan-out, /NT mean
}

// ---------------------------------------------------------------- launch
extern "C" void kernel_launch(void* const* d_in, const int* in_sizes, int n_in,
                              void* d_out, int out_size, void* d_ws, size_t ws_size,
                              hipStream_t stream) {
    const float* pred   = (const float*)d_in[0];
    const float* target = (const float*)d_in[1];
    float* out = (float*)d_out;

    float*    r_bg = (float*)d_ws;              // B*H*W floats, later aliased as w
    float*    r_fg = r_bg + NT_;                // B*H*W floats
    unsigned* mm   = (unsigned*)(r_fg + NT_);   // 2*B_ uints (min bits | max bits)
    float*    mmf  = (float*)(mm + 2 * B_);     // 2*B_ floats (wmin | inv_denom)
    float*    part = mmf + 2 * B_;              // NWAVE per-wave partials
    float*    wmap = r_bg;                      // safe alias

    init_mm_kernel<<<1, 64, 0, stream>>>(mm);
    row_minplus_kernel<<<B_ * H_, 128, 0, stream>>>(target, r_bg, r_fg);
    col_minplus_w_kernel<<<B_ * (W_ / TC), 256, 0, stream>>>(r_bg, r_fg, wmap, mm);
    finalize_mm_kernel<<<1, 64, 0, stream>>>(mm, mmf);
    loss_wmma_kernel<<<NWAVE / 8, 256, 0, stream>>>(pred, target, wmap, mmf, part);
    final_kernel<<<1, 256, 0, stream>>>(part, out);
}